// GraphConvolutionalLayer_68272800137826
// MI455X (gfx1250) — compile-verified
//
#include <hip/hip_runtime.h>

typedef float v2f __attribute__((ext_vector_type(2)));
typedef float v8f __attribute__((ext_vector_type(8)));

#define CH 128
#define LDS_PITCH 130  // pad 16x128 tile rows: stride 130 mod 64 = 2 -> 16 lanes hit distinct banks

// ---------------- kernel 1: zero the workspace accumulators ----------------
__global__ void gcn_zero_ws(float* __restrict__ agg, float* __restrict__ deg, int n_nodes) {
    int i = blockIdx.x * blockDim.x + threadIdx.x;
    int total = n_nodes * CH;
    if (i < total) agg[i] = 0.0f;
    if (i < n_nodes) deg[i] = 0.0f;
}

// ---------------- kernel 2: edge scatter (one wave32 per edge) ----------------
// Edge (a,b): node a accumulates x[b], node b accumulates x[a]; degrees +1 each.
__global__ __launch_bounds__(256)
void gcn_scatter(const float* __restrict__ x, const int* __restrict__ ei,
                 float* __restrict__ agg, float* __restrict__ deg, int n_edges) {
    int wave = (blockIdx.x * blockDim.x + threadIdx.x) >> 5;
    int lane = threadIdx.x & 31;
    if (wave >= n_edges) return;

    int na = ei[2 * wave + 0];
    int nb = ei[2 * wave + 1];

    // coalesced float4 gather: 32 lanes x 16B = full 128-ch row
    float4 va = *(const float4*)(x + (size_t)na * CH + lane * 4);
    float4 vb = *(const float4*)(x + (size_t)nb * CH + lane * 4);

    float* da = agg + (size_t)na * CH + lane * 4;
    float* db = agg + (size_t)nb * CH + lane * 4;
    atomicAdd(da + 0, vb.x); atomicAdd(da + 1, vb.y);
    atomicAdd(da + 2, vb.z); atomicAdd(da + 3, vb.w);
    atomicAdd(db + 0, va.x); atomicAdd(db + 1, va.y);
    atomicAdd(db + 2, va.z); atomicAdd(db + 3, va.w);

    if (lane == 0) {
        atomicAdd(deg + na, 1.0f);
        atomicAdd(deg + nb, 1.0f);
    }
}

// ---------------- kernel 3: fused dual-GEMM + ReLU via V_WMMA_F32_16X16X4_F32 ----
// out[16 rows x 128 cols] = relu( aggNorm(16x128) @ w^T + x(16x128) @ b^T )
// One block = one 16-row node tile; 8 waves each own a 16-col N tile.
__global__ __launch_bounds__(256)
void gcn_gemm_fused(const float* __restrict__ x, const float* __restrict__ w,
                    const float* __restrict__ bias, const float* __restrict__ agg,
                    const float* __restrict__ deg, float* __restrict__ out, int n_nodes) {
    __shared__ float lds_a[16 * LDS_PITCH];  // degree-normalized agg tile
    __shared__ float lds_x[16 * LDS_PITCH];  // x tile

    const int row0 = blockIdx.x * 16;
    const int t = threadIdx.x;

    // Stage A tiles: 2 passes x 256 threads x float4 covers 16x128 floats per tile.
    #pragma unroll
    for (int p = 0; p < 2; ++p) {
        int idx = t + p * 256;
        int r = idx >> 5;              // 0..15
        int c = (idx & 31) << 2;       // 0,4,...,124
        int grow = row0 + r;
        float d = deg[grow];
        float scale = d > 0.0f ? 1.0f / d : 0.0f;  // mean w/ zero fallback
        float4 av = *(const float4*)(agg + (size_t)grow * CH + c);
        float4 xv = *(const float4*)(x + (size_t)grow * CH + c);
        float* la = &lds_a[r * LDS_PITCH + c];
        float* lx = &lds_x[r * LDS_PITCH + c];
        la[0] = av.x * scale; la[1] = av.y * scale;
        la[2] = av.z * scale; la[3] = av.w * scale;
        lx[0] = xv.x; lx[1] = xv.y; lx[2] = xv.z; lx[3] = xv.w;
    }
    __syncthreads();

    const int wv   = t >> 5;           // wave id 0..7 -> N tile
    const int lane = t & 31;
    const int n0   = wv * 16;
    const int m    = lane & 15;        // A row (lanes 0-15) / shared N col index
    const int khi  = (lane >> 4) << 1; // lanes 16-31 hold K+2,K+3 halves
    const int n    = n0 + m;           // this lane's output column

    const float* wrow = w    + (size_t)n * CH;  // w^T column n == w row n
    const float* brow = bias + (size_t)n * CH;

    v8f acc = {};

    // Phase 1: aggNorm @ w^T  (K = 128, 32 WMMA steps)
    #pragma unroll
    for (int kk = 0; kk < CH; kk += 4) {
        v2f af = *(const v2f*)&lds_a[m * LDS_PITCH + kk + khi];
        v2f bf = *(const v2f*)&wrow[kk + khi];
        acc = __builtin_amdgcn_wmma_f32_16x16x4_f32(
            false, af, false, bf, (short)0, acc, false, false);
    }
    // Phase 2: x @ b^T  (accumulate into same C)
    #pragma unroll
    for (int kk = 0; kk < CH; kk += 4) {
        v2f af = *(const v2f*)&lds_x[m * LDS_PITCH + kk + khi];
        v2f bf = *(const v2f*)&brow[kk + khi];
        acc = __builtin_amdgcn_wmma_f32_16x16x4_f32(
            false, af, false, bf, (short)0, acc, false, false);
    }

    // C layout: VGPR v -> M = v (lanes 0-15) / v+8 (lanes 16-31); N = lane & 15
    const int mbase = (lane >> 4) ? 8 : 0;
    #pragma unroll
    for (int v = 0; v < 8; ++v) {
        int grow = row0 + mbase + v;
        float val = acc[v];
        out[(size_t)grow * CH + n] = val > 0.0f ? val : 0.0f;
    }
}

extern "C" void kernel_launch(void* const* d_in, const int* in_sizes, int n_in,
                              void* d_out, int out_size, void* d_ws, size_t ws_size,
                              hipStream_t stream) {
    const float* x   = (const float*)d_in[0];
    const int*   ei  = (const int*)d_in[1];
    const float* w   = (const float*)d_in[2];
    const float* b   = (const float*)d_in[3];
    float*       out = (float*)d_out;

    const int n_nodes = in_sizes[0] / CH;
    const int n_edges = in_sizes[1] / 2;

    float* agg = (float*)d_ws;                       // n_nodes*CH floats
    float* deg = agg + (size_t)n_nodes * CH;         // n_nodes floats

    // 1) zero accumulators
    {
        int total = n_nodes * CH;
        int blocks = (total + 255) / 256;
        gcn_zero_ws<<<blocks, 256, 0, stream>>>(agg, deg, n_nodes);
    }
    // 2) edge scatter: 8 edges (waves) per 256-thread block
    {
        int blocks = (n_edges + 7) / 8;
        gcn_scatter<<<blocks, 256, 0, stream>>>(x, ei, agg, deg, n_edges);
    }
    // 3) fused dual-GEMM + ReLU: one block per 16 node rows
    {
        int blocks = (n_nodes + 15) / 16;
        gcn_gemm_fused<<<blocks, 256, 0, stream>>>(x, w, b, agg, deg, out, n_nodes);
    }
}